// CoxPHLoss_62878321214164
// MI455X (gfx1250) — compile-verified
//
#include <hip/hip_runtime.h>
#include <cstdint>

// CoxPHLoss (Efron) for MI455X / gfx1250.
// B=16, N=32768, E=8, durations in [0,1000).
// One workgroup (1024 threads, 32 wave32s) per (b,e) risk set.
// Counting-bucket reformulation of sort + cumlogsumexp + tie-group math:
//   two async-to-LDS streamed passes over N, then O(1000) bucket finalize.
// Depth-2 async pipeline (3 rotating LDS tile buffers) to hide L2/HBM latency.

#define NBUCKET 1000
#define BLOCK   1024
#define TILE    1024
#define NBUF    3
#define NN      32768
#define EE      8
#define NITER   (NN / TILE)   // 32

// ---- CDNA5 async-tensor path: global -> LDS DMA, tracked by ASYNCcnt ----
__device__ __forceinline__ void async_ld_b32(uint32_t lds_addr, uint64_t gaddr) {
  // GV mode: per-lane 64-bit global address, per-lane 32-bit LDS dest address.
  asm volatile("global_load_async_to_lds_b32 %0, %1, off"
               :: "v"(lds_addr), "v"(gaddr)
               : "memory");
}
#define WAIT_ASYNC_6() asm volatile("s_wait_asynccnt 0x6" ::: "memory")
#define WAIT_ASYNC_3() asm volatile("s_wait_asynccnt 0x3" ::: "memory")
#define WAIT_ASYNC_0() asm volatile("s_wait_asynccnt 0x0" ::: "memory")

__device__ __forceinline__ uint32_t lds_addr_of(const void* p) {
  // LDS aperture: flat addr low 32 bits are the LDS byte address.
  return (uint32_t)(uintptr_t)p;
}

__global__ __launch_bounds__(BLOCK)
void cox_riskset_kernel(const float* __restrict__ logh,
                        const int*   __restrict__ events,
                        const int*   __restrict__ durations,
                        float* __restrict__ g_sum, int* __restrict__ g_cnt) {
  __shared__ int    s_cnt[NBUCKET];   // events per duration
  __shared__ int    s_idx[NBUCKET];   // max original index among events (tie-group end)
  __shared__ float  s_all[NBUCKET];   // sum exp(lh-Mg), all samples
  __shared__ float  s_le [NBUCKET];   // sum exp(lh-Mg), samples with idx <= s_idx[d]
  __shared__ float  s_ev [NBUCKET];   // sum exp(lh-Mg), events only
  __shared__ float  s_suf[NBUCKET];   // suffix sum of s_all over d' > d
  __shared__ float  s_tF[NBUF][TILE]; // async-staged logh tiles
  __shared__ int    s_tE[NBUF][TILE]; // async-staged events tiles
  __shared__ int    s_tD[NBUF][TILE]; // async-staged durations tiles
  __shared__ float  s_wmax[BLOCK / 32];
  __shared__ double s_logden;
  __shared__ double s_evlh;
  __shared__ int    s_nev;
  __shared__ float  s_Mg;

  const int tid = threadIdx.x;
  const int b   = blockIdx.x / EE;
  const int e   = blockIdx.x % EE;

  // Column base pointers for this risk set (stride EE between samples).
  const size_t colbase = (size_t)b * NN * EE + (size_t)e;
  const float* pF = logh      + colbase;
  const int*   pE = events    + colbase;
  const int*   pD = durations + colbase;

  for (int d = tid; d < NBUCKET; d += BLOCK) {
    s_cnt[d] = 0; s_idx[d] = -1;
    s_all[d] = 0.0f; s_le[d] = 0.0f; s_ev[d] = 0.0f;
  }
  if (tid == 0) { s_logden = 0.0; s_evlh = 0.0; s_nev = 0; }
  __syncthreads();

  auto issue = [&](int t) {
    const int buf = t % NBUF;
    const size_t off = (size_t)(t * TILE + tid) * EE;
    async_ld_b32(lds_addr_of(&s_tF[buf][tid]), (uint64_t)(uintptr_t)(pF + off));
    async_ld_b32(lds_addr_of(&s_tE[buf][tid]), (uint64_t)(uintptr_t)(pE + off));
    async_ld_b32(lds_addr_of(&s_tD[buf][tid]), (uint64_t)(uintptr_t)(pD + off));
  };

  // ---------------- Pass 1: event counts, tie-group ends, global max ----------------
  float  lm   = -__builtin_inff();
  double evlh = 0.0;
  int    nev  = 0;

  issue(0); issue(1);
  for (int t = 0; t < NITER; ++t) {
    if      (t + 2 < NITER) { issue(t + 2); WAIT_ASYNC_6(); }
    else if (t + 1 < NITER) { WAIT_ASYNC_3(); }
    else                    { WAIT_ASYNC_0(); }
    const int   buf = t % NBUF;
    const float lh  = s_tF[buf][tid];
    const int   ev  = s_tE[buf][tid];
    const int   d   = s_tD[buf][tid];
    lm = fmaxf(lm, lh);
    if (ev != 0) {
      atomicAdd(&s_cnt[d], 1);
      atomicMax(&s_idx[d], t * TILE + tid);
      evlh += (double)lh;
      ++nev;
    }
  }

  // Block max of lh (wave32 shuffle tree, then cross-wave via LDS).
  for (int o = 16; o > 0; o >>= 1) lm = fmaxf(lm, __shfl_xor(lm, o, 32));
  const int lane = tid & 31, wid = tid >> 5;
  if (lane == 0) s_wmax[wid] = lm;
  atomicAdd(&s_nev, nev);
  atomicAdd(&s_evlh, evlh);
  __syncthreads();
  if (tid < (BLOCK / 32)) {
    float v = s_wmax[tid];
    for (int o = 16; o > 0; o >>= 1) v = fmaxf(v, __shfl_xor(v, o, 32));
    if (tid == 0) s_Mg = v;
  }
  __syncthreads();
  const float Mg = s_Mg;

  // ---------------- Pass 2: bucket exp-sums (all / prefix-to-last-event / events) ----
  issue(0); issue(1);
  for (int t = 0; t < NITER; ++t) {
    if      (t + 2 < NITER) { issue(t + 2); WAIT_ASYNC_6(); }
    else if (t + 1 < NITER) { WAIT_ASYNC_3(); }
    else                    { WAIT_ASYNC_0(); }
    const int   buf = t % NBUF;
    const float lh  = s_tF[buf][tid];
    const int   ev  = s_tE[buf][tid];
    const int   d   = s_tD[buf][tid];
    const float x   = __expf(lh - Mg);
    atomicAdd(&s_all[d], x);
    if (ev != 0) atomicAdd(&s_ev[d], x);
    if ((t * TILE + tid) <= s_idx[d]) atomicAdd(&s_le[d], x);  // s_idx = -1 -> skipped
  }
  __syncthreads();

  // Suffix sums over durations (descending order == stable-sort order).
  if (tid == 0) {
    float acc = 0.0f;
    for (int d = NBUCKET - 1; d >= 0; --d) { s_suf[d] = acc; acc += s_all[d]; }
  }
  __syncthreads();

  // ---------------- Finalize: Efron terms per tie group ----------------
  // etl = Mg + log(R), R = sum exp over {dur > d} U {dur == d, idx <= idx_e[d]}
  // ls(m) = etl + log1p(-(m/tc) * T / R),  T = tie-group event exp-sum.
  double gl = 0.0;
  for (int d = tid; d < NBUCKET; d += BLOCK) {
    const int tc = s_cnt[d];
    if (tc > 0) {
      const float R    = s_suf[d] + s_le[d];
      const float T    = s_ev[d];
      const float etl  = __logf(R) + Mg;
      const float finv = T / ((float)tc * R);
      for (int m = 0; m < tc; ++m)
        gl += (double)(etl + log1pf(-(float)m * finv));
    }
  }
  atomicAdd(&s_logden, gl);
  __syncthreads();

  if (tid == 0) {
    const double nevf = (s_nev > 0) ? (double)s_nev : 1.0;
    const double per  = (s_logden - s_evlh) / nevf;   // log_den - log_num
    if (per > 0.0) {
      atomicAdd(g_sum, (float)per);
      atomicAdd(g_cnt, 1);
    }
  }
}

__global__ void cox_init_kernel(float* g_sum, int* g_cnt) {
  *g_sum = 0.0f;
  *g_cnt = 0;
}

__global__ void cox_fin_kernel(float* out, const float* g_sum, const int* g_cnt) {
  out[0] = g_sum[0] / (float)g_cnt[0];
}

extern "C" void kernel_launch(void* const* d_in, const int* in_sizes, int n_in,
                              void* d_out, int out_size, void* d_ws, size_t ws_size,
                              hipStream_t stream) {
  const float* logh      = (const float*)d_in[0];
  const int*   events    = (const int*)d_in[1];
  const int*   durations = (const int*)d_in[2];
  float* out   = (float*)d_out;
  float* g_sum = (float*)d_ws;
  int*   g_cnt = (int*)((char*)d_ws + 16);

  cox_init_kernel<<<1, 1, 0, stream>>>(g_sum, g_cnt);
  cox_riskset_kernel<<<16 * EE, BLOCK, 0, stream>>>(logh, events, durations, g_sum, g_cnt);
  cox_fin_kernel<<<1, 1, 0, stream>>>(out, g_sum, g_cnt);
}